// GuidedFilter_34196529610978
// MI455X (gfx1250) — compile-verified
//
#include <hip/hip_runtime.h>

typedef __attribute__((ext_vector_type(2))) float v2f;
typedef __attribute__((ext_vector_type(8))) float v8f;

#define HW    2048
#define IMG   (HW * HW)
#define KH    17
#define PC    8           // KH/2
#define LDSW  24          // LDS row stride, supports KW up to 9
#define TROWS 32          // 16 output rows + 2*PC halo
#define WPROWS 48         // zero-padded banded weight rows: t+16 for t in [-16, 31]

// One pass: Xout = Xin + conv(Yin - Xin, Wk) / Nplane   (cross-correlation, zero pad)
// One 16x16 output tile per wave; 8 waves per 256-thread block.
// Vertical 17-tap stencil expressed as banded matmul via V_WMMA_F32_16X16X4_F32.
template <int KW>
__global__ __launch_bounds__(256) void guided_pass(
    const float* __restrict__ Xin, const float* __restrict__ Yin,
    const float* __restrict__ Wk,  const float* __restrict__ Nplane,
    float* __restrict__ Xout)
{
    constexpr int PR = KW / 2;
    constexpr int TW = 16 + KW - 1;

    __shared__ float wpad[WPROWS * KW];     // zero-padded band weights: row (t+16)
    __shared__ float tile[8][TROWS * LDSW]; // per-wave staged (y - x) patch

    const int lane = threadIdx.x & 31;
    const int wid  = threadIdx.x >> 5;

    // Stage zero-padded weights in one race-free pass (rows 16..32 hold the taps)
    for (int i = threadIdx.x; i < WPROWS * KW; i += 256) {
        int j = i - 16 * KW;
        wpad[i] = (j >= 0 && j < KH * KW) ? Wk[j] : 0.0f;
    }

    // Tile coordinates: t enumerates batch * 128 * 128 tiles
    const int TPR = HW / 16;            // 128
    const int TPI = TPR * TPR;
    const int t   = blockIdx.x * 8 + wid;
    const int img = t / TPI;
    const int rem = t % TPI;
    const int row0 = (rem / TPR) * 16;
    const int col0 = (rem % TPR) * 16;

    const float* Xi = Xin + (size_t)img * IMG;
    const float* Yi = Yin + (size_t)img * IMG;
    float*       Xo = Xout + (size_t)img * IMG;

    // Stage D = y - x patch: rows [row0-8, row0+23], cols [col0-PR, col0+15+PR]
    float* my = tile[wid];
    for (int idx = lane; idx < TROWS * TW; idx += 32) {
        int r = idx / TW, c = idx % TW;
        int gr = row0 + r - PC;
        int gc = col0 + c - PR;
        float d = 0.0f;
        if (gr >= 0 && gr < HW && gc >= 0 && gc < HW) {
            size_t o = (size_t)gr * HW + gc;
            d = Yi[o] - Xi[o];
        }
        my[r * LDSW + c] = d;
    }
    __syncthreads();

    // Banded matmul: O(16x16) = sum_kx A_kx(16x32) * B_kx(32x16), K split into 8 x (K=4)
    // A layout (32-bit, 16x4): lanes 0-15 hold M=lane, K={j0, j0+1}; lanes 16-31 K={j0+2, j0+3}
    // B layout (32-bit, 4x16): lanes 0-15 hold K rows {j0, j0+1}; lanes 16-31 rows {j0+2, j0+3}
    v8f acc = {};
    const int half = lane >> 4;
    const int Mrow = lane & 15;
    const int n    = lane & 15;

#pragma unroll
    for (int s = 0; s < 8; ++s) {
        const int jA   = 4 * s + 2 * half;          // patch-row (K index) for vgpr .x
        const int tpad = jA - Mrow + 16;            // banded weight row (pre-zero-padded)
        const float* wrA = &wpad[tpad * KW];        // tap row for .x
        const float* wrB = wrA + KW;                // tap row for .y (t+1)
        const float* ld0 = &my[jA * LDSW + n];      // patch row jA
        const float* ld1 = ld0 + LDSW;              // patch row jA+1

#pragma unroll
        for (int kx = 0; kx < KW; ++kx) {
            v2f a, b;
            a.x = wrA[kx];
            a.y = wrB[kx];
            b.x = ld0[kx];
            b.y = ld1[kx];
            acc = __builtin_amdgcn_wmma_f32_16x16x4_f32(
                false, a, false, b, (short)0, acc, false, false);
        }
    }

    // Epilogue: X_out = X_in + acc / N   (C/D layout: vgpr v -> row v + 8*half, col n)
#pragma unroll
    for (int v = 0; v < 8; ++v) {
        int row = row0 + v + 8 * half;
        int col = col0 + n;
        size_t o = (size_t)row * HW + col;
        Xo[o] = Xi[o] + acc[v] / Nplane[o];
    }
}

template <int KW>
static void run_three_passes(const float* X, const float* y, const float* k,
                             const float* Nn, float* out, float* tmp,
                             int batch, hipStream_t stream)
{
    const int tiles = batch * (HW / 16) * (HW / 16);
    dim3 grid(tiles / 8), block(256);
    // Pass 0: X0 (d_in) -> d_out
    guided_pass<KW><<<grid, block, 0, stream>>>(X,   y, k,               Nn,           out);
    // Pass 1: X1 (d_out) -> tmp
    guided_pass<KW><<<grid, block, 0, stream>>>(out, y, k + KH * KW,     Nn + IMG,     tmp);
    // Pass 2: X2 (tmp) -> d_out
    guided_pass<KW><<<grid, block, 0, stream>>>(tmp, y, k + 2 * KH * KW, Nn + 2 * IMG, out);
}

extern "C" void kernel_launch(void* const* d_in, const int* in_sizes, int n_in,
                              void* d_out, int out_size, void* d_ws, size_t ws_size,
                              hipStream_t stream) {
    const float* X  = (const float*)d_in[0];   // (2,1,2048,2048) f32
    const float* y  = (const float*)d_in[1];   // (2,1,2048,2048) f32
    const float* k  = (const float*)d_in[2];   // (3,1,17,KW)     f32
    const float* Nn = (const float*)d_in[3];   // (3,1,2048,2048) f32
    float* out = (float*)d_out;
    float* tmp = (float*)d_ws;

    const int KW    = in_sizes[2] / (3 * KH);  // = 5 for RX=8, angles ±10
    const int batch = in_sizes[0] / IMG;       // = 2

    switch (KW) {
        case 1: run_three_passes<1>(X, y, k, Nn, out, tmp, batch, stream); break;
        case 3: run_three_passes<3>(X, y, k, Nn, out, tmp, batch, stream); break;
        case 7: run_three_passes<7>(X, y, k, Nn, out, tmp, batch, stream); break;
        case 9: run_three_passes<9>(X, y, k, Nn, out, tmp, batch, stream); break;
        case 5:
        default: run_three_passes<5>(X, y, k, Nn, out, tmp, batch, stream); break;
    }
}